// PeakDetector_8263517077714
// MI455X (gfx1250) — compile-verified
//
#include <hip/hip_runtime.h>
#include <stdint.h>

#define N        2048
#define BLOCK    256
#define EPT      (N / BLOCK)   // 8 elements per thread
#define MIN_DIST 10
#define HEIGHTF  0.1f
#define PROMF    0.05f

// Low 32 bits of a flat (generic) shared-memory address are the wave-relative
// LDS byte offset (ISA 10.2: LDS aperture keeps the low 32 address bits).
// Confirmed by round-1 codegen (async dst folded to literal LDS offsets).
__device__ __forceinline__ uint32_t lds_byte_off(const void* p) {
    return (uint32_t)(uintptr_t)p;
}

__global__ __launch_bounds__(BLOCK)
void PeakDetector_kernel(const float* __restrict__ x, float* __restrict__ out, int rows) {
    __shared__ __align__(16) float sx[N];        // raw row, then normalized in place
    __shared__ unsigned long long  keys[N];      // (f32 bits << 11) | (2047 - i)
    __shared__ float               redmn[BLOCK];
    __shared__ float               redmx[BLOCK];
    __shared__ unsigned char       cand[N];
    __shared__ unsigned char       sup[N];
    __shared__ unsigned char       keepm[N];

    const int tid = threadIdx.x;
    const int row = blockIdx.x;
    if (row >= rows) return;

    const uint64_t gbase = (uint64_t)(uintptr_t)(x + (size_t)row * N);

    // ---- 1) DMA row into LDS: gfx1250 async global->LDS, 512 B/instruction ----
    //      (32 lanes x 16 B per GLOBAL_LOAD_ASYNC_TO_LDS_B128; ASYNCcnt-tracked)
    #pragma unroll
    for (int k = 0; k < EPT / 4; ++k) {                // 2 issues per lane
        const int e = (k * BLOCK + tid) * 4;           // float index, 16B-aligned
        uint32_t dst = lds_byte_off(&sx[e]);
        uint32_t off = (uint32_t)(e * 4);              // byte offset into row
        asm volatile("global_load_async_to_lds_b128 %0, %1, %2"
                     :: "v"(dst), "v"(off), "s"(gbase) : "memory");
    }
    asm volatile("s_wait_asynccnt 0" ::: "memory");    // this wave's copies done
    __syncthreads();                                   // publish LDS across waves

    // ---- 2) row min/max reduction ----
    float mn = 3.402823466e38f, mx = -3.402823466e38f;
    #pragma unroll
    for (int k = 0; k < EPT; ++k) {
        float v = sx[k * BLOCK + tid];
        mn = fminf(mn, v);
        mx = fmaxf(mx, v);
    }
    redmn[tid] = mn; redmx[tid] = mx;
    __syncthreads();
    for (int s = BLOCK / 2; s > 0; s >>= 1) {
        if (tid < s) {
            redmn[tid] = fminf(redmn[tid], redmn[tid + s]);
            redmx[tid] = fmaxf(redmx[tid], redmx[tid + s]);
        }
        __syncthreads();
    }
    mn = redmn[0];
    const float denom = redmx[0] - mn + 1e-5f;         // exact: same op order as ref
    __syncthreads();

    // ---- 3) normalize in place (IEEE divide -> bitwise identical to ref) ----
    #pragma unroll
    for (int k = 0; k < EPT; ++k) {
        const int e = k * BLOCK + tid;
        sx[e] = (sx[e] - mn) / denom;
    }
    __syncthreads();

    // ---- 4) candidate mask + sort keys + init flags ----
    #pragma unroll
    for (int k = 0; k < EPT; ++k) {
        const int e = k * BLOCK + tid;
        const float v = sx[e];
        bool isp = (e > 0) && (e < N - 1) && (v > sx[e - 1]) && (v > sx[e + 1]);
        cand[e]  = (isp && (v >= HEIGHTF)) ? 1 : 0;
        sup[e]   = 0;
        keepm[e] = 0;
        // v >= 0 after min-max norm -> float bits order == numeric order.
        // Low field (2047 - i): for equal values smaller i sorts first in
        // descending key order == stable argsort(-xn).
        keys[e] = ((unsigned long long)__float_as_uint(v) << 11)
                | (unsigned long long)(N - 1 - e);
    }
    __syncthreads();

    // ---- 5) bitonic sort keys, descending ----
    for (unsigned kk = 2; kk <= (unsigned)N; kk <<= 1) {
        for (unsigned j = kk >> 1; j > 0; j >>= 1) {
            for (int t = tid; t < N; t += BLOCK) {
                const unsigned ixj = (unsigned)t ^ j;
                if (ixj > (unsigned)t) {
                    unsigned long long a = keys[t], b = keys[ixj];
                    const bool blk = (((unsigned)t & kk) == 0);
                    if (blk ? (a < b) : (a > b)) { keys[t] = b; keys[ixj] = a; }
                }
            }
            __syncthreads();
        }
    }

    // ---- 6) greedy distance suppression (sequential by construction) ----
    if (tid == 0) {
        for (int s = 0; s < N; ++s) {
            const int idx = (N - 1) - (int)(keys[s] & (unsigned long long)(N - 1));
            if (cand[idx] && !sup[idx]) {
                keepm[idx] = 1;
                int lo = idx - (MIN_DIST - 1); if (lo < 0) lo = 0;
                int hi = idx + (MIN_DIST - 1); if (hi > N - 1) hi = N - 1;
                for (int d = lo; d <= hi; ++d)
                    if (d != idx) sup[d] = 1;
            }
        }
    }
    __syncthreads();

    // ---- 7) prominence for surviving peaks + write output ----
    #pragma unroll 1
    for (int k = 0; k < EPT; ++k) {
        const int e = k * BLOCK + tid;
        float r = 0.0f;
        if (keepm[e]) {
            const float v = sx[e];
            float lmin = v;                       // min over (L, i], includes i
            for (int j = e - 1; j >= 0; --j) {
                const float u = sx[j];
                if (u > v) break;                 // L excluded
                lmin = fminf(lmin, u);
            }
            float rmin = v;                       // min over [i, R)
            for (int j = e + 1; j < N; ++j) {
                const float u = sx[j];
                if (u > v) break;                 // R excluded
                rmin = fminf(rmin, u);
            }
            const float prom = v - fmaxf(lmin, rmin);
            r = (prom >= PROMF) ? 1.0f : 0.0f;
        }
        out[(size_t)row * N + e] = r;
    }
}

extern "C" void kernel_launch(void* const* d_in, const int* in_sizes, int n_in,
                              void* d_out, int out_size, void* d_ws, size_t ws_size,
                              hipStream_t stream) {
    (void)n_in; (void)d_ws; (void)ws_size; (void)out_size;
    const float* x  = (const float*)d_in[0];
    float* out      = (float*)d_out;
    const int rows  = in_sizes[0] / N;      // 16 for the reference shapes
    PeakDetector_kernel<<<dim3(rows), dim3(BLOCK), 0, stream>>>(x, out, rows);
}